// DLinearC_24464133718182
// MI455X (gfx1250) — compile-verified
//
#include <hip/hip_runtime.h>
#include <hip/hip_bf16.h>
#include <stdint.h>

// Problem constants
#define SEQ   512
#define PRED  720
#define NB    32
#define NC    512
#define NE    8
#define NR    (NB*NC)          // 16384 rows (b,c)

typedef __attribute__((ext_vector_type(2))) float v2f;
typedef __attribute__((ext_vector_type(8))) float v8f;

// workspace layout (floats)
#define WS_SEA   ((size_t)0)
#define WS_TREND ((size_t)NR*SEQ)
#define WS_WFS   ((size_t)2*NR*SEQ)
#define WS_WFT   ((size_t)2*NR*SEQ + (size_t)NR*NE)
#define WS_GATE  ((size_t)2*NR*SEQ + (size_t)2*NR*NE)

// CDNA5 async global->LDS path (ASYNCcnt), with sync fallback
#if __has_builtin(__builtin_amdgcn_global_load_async_to_lds_b128) && \
    __has_builtin(__builtin_amdgcn_s_wait_asynccnt)
#define HAS_ASYNC 1
#else
#define HAS_ASYNC 0
#endif

typedef int v4i_t __attribute__((vector_size(16)));
typedef __attribute__((address_space(1))) v4i_t* g128_t;   // global int4*
typedef __attribute__((address_space(3))) v4i_t* l128_t;   // LDS    int4*

__device__ __forceinline__ void copy16_to_lds(const float* g, float* l) {
#if HAS_ASYNC
    __builtin_amdgcn_global_load_async_to_lds_b128(
        (g128_t)(uintptr_t)g, (l128_t)(uintptr_t)l, 0, 0);
#else
    *(float4*)l = *(const float4*)g;
#endif
}

__device__ __forceinline__ void copy_wait() {
#if HAS_ASYNC
    __builtin_amdgcn_s_wait_asynccnt(0);
#endif
}

// ---------------------------------------------------------------------------
// Kernel 1: series decomposition (moving average k=25, replicate padding)
// x: [B,S,C] -> sea/trend in [B,S,C] layout (coalesced along C)
// ---------------------------------------------------------------------------
__global__ __launch_bounds__(256) void k_decompose(const float* __restrict__ x,
                                                   float* __restrict__ sea,
                                                   float* __restrict__ trend) {
    int c = blockIdx.x * 256 + threadIdx.x;
    int b = blockIdx.y;
    const float* xp = x     + (size_t)b * SEQ * NC + c;
    float*       sp = sea   + (size_t)b * SEQ * NC + c;
    float*       tp = trend + (size_t)b * SEQ * NC + c;

    float wsum = 12.0f * xp[0];               // 12 replicated left-pads
    #pragma unroll
    for (int j = 0; j <= 12; ++j) wsum += xp[(size_t)j * NC];
    const float inv25 = 1.0f / 25.0f;
    for (int s = 0; s < SEQ; ++s) {
        float tr = wsum * inv25;
        float v  = xp[(size_t)s * NC];
        tp[(size_t)s * NC] = tr;
        sp[(size_t)s * NC] = v - tr;
        int ia = s + 13; if (ia > SEQ - 1) ia = SEQ - 1;
        int ir = s - 12; if (ir < 0)       ir = 0;
        wsum += xp[(size_t)ia * NC] - xp[(size_t)ir * NC];
    }
}

// ---------------------------------------------------------------------------
// Kernel 2: gating (logits over 8 experts, softmax, top-2 -> sparse combine
// weights). Also stores full trend gate probs for the mean output.
// ---------------------------------------------------------------------------
__device__ __forceinline__ void softmax_top2(const float* l, float* p, float* wf) {
    float m = l[0];
    #pragma unroll
    for (int e = 1; e < NE; ++e) m = fmaxf(m, l[e]);
    float sum = 0.0f;
    #pragma unroll
    for (int e = 0; e < NE; ++e) { p[e] = __expf(l[e] - m); sum += p[e]; }
    float inv = 1.0f / sum;
    #pragma unroll
    for (int e = 0; e < NE; ++e) p[e] *= inv;
    int i1 = 0;
    #pragma unroll
    for (int e = 1; e < NE; ++e) if (p[e] > p[i1]) i1 = e;
    int i2 = (i1 == 0) ? 1 : 0;
    #pragma unroll
    for (int e = 0; e < NE; ++e) if (e != i1 && p[e] > p[i2]) i2 = e;
    #pragma unroll
    for (int e = 0; e < NE; ++e) wf[e] = (e == i1 || e == i2) ? p[e] : 0.0f;
}

__global__ __launch_bounds__(256) void k_gate(const float* __restrict__ sea,
                                              const float* __restrict__ trend,
                                              const float* __restrict__ WgS,
                                              const float* __restrict__ WgT,
                                              float* __restrict__ wfS,
                                              float* __restrict__ wfT,
                                              float* __restrict__ gate) {
    __shared__ float wg[2 * NE * SEQ];   // 32 KB
    int t = threadIdx.x;
    #pragma unroll
    for (int i = 0; i < 16; ++i) {
        int idx = t + i * 256;           // 0..4095
        wg[idx]            = WgS[idx];
        wg[NE * SEQ + idx] = WgT[idx];
    }
    __syncthreads();

    int c = blockIdx.x * 256 + t;
    int b = blockIdx.y;
    const float* sp = sea   + (size_t)b * SEQ * NC + c;
    const float* tp = trend + (size_t)b * SEQ * NC + c;

    float ls[NE] = {0}, lt[NE] = {0};
    for (int s = 0; s < SEQ; ++s) {
        float vs = sp[(size_t)s * NC];
        float vt = tp[(size_t)s * NC];
        #pragma unroll
        for (int e = 0; e < NE; ++e) {
            ls[e] = fmaf(vs, wg[e * SEQ + s],            ls[e]);
            lt[e] = fmaf(vt, wg[NE * SEQ + e * SEQ + s], lt[e]);
        }
    }
    size_t r = (size_t)b * NC + c;
    float ps[NE], pt[NE], ws[NE], wt[NE];
    softmax_top2(ls, ps, ws);
    softmax_top2(lt, pt, wt);
    #pragma unroll
    for (int e = 0; e < NE; ++e) {
        wfS[r * NE + e]  = ws[e];
        wfT[r * NE + e]  = wt[e];
        gate[r * NE + e] = pt[e];
    }
}

// ---------------------------------------------------------------------------
// Kernel 3: gate_trend.mean(axis=0) -> out tail [C,E] (deterministic reduce)
// ---------------------------------------------------------------------------
__global__ __launch_bounds__(256) void k_gatemean(const float* __restrict__ gate,
                                                  float* __restrict__ outg) {
    int t = blockIdx.x * 256 + threadIdx.x;   // 0..4095 == c*8+e
    int c = t >> 3, e = t & 7;
    float s = 0.0f;
    #pragma unroll
    for (int b = 0; b < NB; ++b) s += gate[((size_t)b * NC + c) * NE + e];
    outg[t] = s * (1.0f / NB);
}

// ---------------------------------------------------------------------------
// Kernel 4: fused MoE GEMM with f32 WMMA, double-buffered LDS staged via
// async global->LDS (ASYNCcnt) copies.
// out[r,p] = sum_{src,e} wfull[src][r,e]*(z_src[r,:].We_src[e,p,:] + be_src[e,p])
// WG tile: 128 rows x 48 cols; 8 waves, wave = 16 rows x 48 cols (3 acc tiles).
// K loop = concat(src, e, s) staged per 16-K chunk; A rows scaled per lane by
// the sparse combine weight. Bias folded in as a K=16 WMMA GEMM.
// ---------------------------------------------------------------------------
#define KC    16
#define ACP   136    // A chunk row stride: [k16][c128+pad8] (b128 stores, conflict-free reads)
#define BPAD  20     // B tile K stride  (16B-aligned b128 stores)
#define APAD  18     // be tile K stride (even -> b64 reads)
#define OPAD  130    // output stage row stride
#define ASZ   (KC*ACP)          // 2176 floats per A buffer
#define BSZ   (NE*48*BPAD)      // 7680 floats per B buffer
#define NCHUNK (2*(SEQ/KC))     // 64 chunks over concat(src, s)

__global__ __launch_bounds__(256) void k_moe(const float* __restrict__ a_sea,
                                             const float* __restrict__ a_trend,
                                             const float* __restrict__ WeS,
                                             const float* __restrict__ WeT,
                                             const float* __restrict__ beS,
                                             const float* __restrict__ beT,
                                             const float* __restrict__ wfS,
                                             const float* __restrict__ wfT,
                                             float* __restrict__ out) {
    __shared__ float sA [2 * ASZ];           // 17.4 KB  double-buffered A
    __shared__ float sB [2 * BSZ];           // 61.4 KB  double-buffered B
                                             // (buffer 0 reused as out stage)
    __shared__ float sBe[48 * APAD];         //  3.5 KB  bias tile [n48][k16]

    int t    = threadIdx.x;
    int lane = t & 31, wave = t >> 5;
    int hi   = lane >> 4, l15 = lane & 15;
    int p0   = blockIdx.x * 48;
    int r0   = blockIdx.y * 128;
    int b    = r0 >> 9;                      // 512 rows per batch
    int c0   = r0 & (NC - 1);
    int arow = wave * 16 + l15;              // this lane's A/scale row in [0,128)

    // per-lane sparse combine weights for row r0+arow (16 = 2 src x 8 experts)
    float wreg[16];
    {
        const float* w0 = wfS + (size_t)(r0 + arow) * NE;
        const float* w1 = wfT + (size_t)(r0 + arow) * NE;
        #pragma unroll
        for (int e = 0; e < NE; ++e) { wreg[e] = w0[e]; wreg[8 + e] = w1[e]; }
    }
    // stage bias tile: k = src*8+e (16), n = 48
    #pragma unroll
    for (int i = 0; i < 3; ++i) {
        int idx = t + i * 256;               // 0..767
        int n = idx % 48, k = idx / 48;
        const float* bep = (k >= 8) ? beT : beS;
        sBe[n * APAD + k] = bep[(k & 7) * PRED + p0 + n];
    }

    // chunk q in [0,64): src = q>>5, k0 = (q&31)*16.  Issue copies into buf.
    auto issue = [&](int q, int buf) {
        const float* Asrc = (q & 32) ? a_trend : a_sea;
        const float* Wsrc = (q & 32) ? WeT : WeS;
        int k0 = (q & 31) * KC;
        // A chunk: 16 k x 128 c, b128 along c (2 per thread)
        #pragma unroll
        for (int i = 0; i < 2; ++i) {
            int idx = t + i * 256;           // 0..511
            int k = idx >> 5, c4 = (idx & 31) * 4;
            copy16_to_lds(Asrc + ((size_t)(b * SEQ + k0 + k)) * NC + c0 + c4,
                          &sA[buf * ASZ + k * ACP + c4]);
        }
        // B chunk: 8 experts x 48 p x 16 k, b128 along k (6 per thread)
        #pragma unroll
        for (int i = 0; i < 6; ++i) {
            int idx = t + i * 256;           // 0..1535
            int row = idx >> 2;              // (e*48+n) in 0..383
            int kk4 = (idx & 3) * 4;
            int e = row / 48, n = row % 48;
            copy16_to_lds(Wsrc + ((size_t)(e * PRED + p0 + n)) * SEQ + k0 + kk4,
                          &sB[buf * BSZ + row * BPAD + kk4]);
        }
    };

    v8f acc0 = {}, acc1 = {}, acc2 = {};

    issue(0, 0);
    #pragma unroll
    for (int src = 0; src < 2; ++src) {      // compile-time src for wreg index
        for (int kc = 0; kc < SEQ / KC; ++kc) {
            const int q   = src * (SEQ / KC) + kc;
            const int buf = q & 1;
            copy_wait();                     // my async copies into buf done
            __syncthreads();                 // everyone's copies visible;
                                             // nobody still reads buf^1
            if (q + 1 < NCHUNK) issue(q + 1, buf ^ 1);

            const float* pA = &sA[buf * ASZ + arow];
            const float* pBb = &sB[buf * BSZ + l15 * BPAD];
            #pragma unroll
            for (int kk = 0; kk < KC; kk += 4) {
                v2f araw;
                araw.x = pA[(kk     + 2 * hi) * ACP];
                araw.y = pA[(kk + 1 + 2 * hi) * ACP];
                v2f as[NE];
                #pragma unroll
                for (int e = 0; e < NE; ++e) {
                    float w = wreg[src * 8 + e];
                    as[e].x = araw.x * w; as[e].y = araw.y * w;
                }
                const float* pB = pBb + kk + 2 * hi;
                #pragma unroll
                for (int e = 0; e < NE; ++e) {
                    v2f b0 = *(const v2f*)(pB + (e * 48 +  0) * BPAD);
                    acc0 = __builtin_amdgcn_wmma_f32_16x16x4_f32(
                        false, as[e], false, b0, (short)0, acc0, false, false);
                }
                #pragma unroll
                for (int e = 0; e < NE; ++e) {
                    v2f b1 = *(const v2f*)(pB + (e * 48 + 16) * BPAD);
                    acc1 = __builtin_amdgcn_wmma_f32_16x16x4_f32(
                        false, as[e], false, b1, (short)0, acc1, false, false);
                }
                #pragma unroll
                for (int e = 0; e < NE; ++e) {
                    v2f b2 = *(const v2f*)(pB + (e * 48 + 32) * BPAD);
                    acc2 = __builtin_amdgcn_wmma_f32_16x16x4_f32(
                        false, as[e], false, b2, (short)0, acc2, false, false);
                }
            }
        }
    }

    // bias as K=16 GEMM: A = wfull (per-row), B = be tile
    #pragma unroll
    for (int kk = 0; kk < 16; kk += 4) {
        v2f a;
        a.x = hi ? wreg[kk + 2] : wreg[kk];
        a.y = hi ? wreg[kk + 3] : wreg[kk + 1];
        const float* Bb = &sBe[l15 * APAD + kk + 2 * hi];
        v2f b0 = *(const v2f*)(Bb);
        acc0 = __builtin_amdgcn_wmma_f32_16x16x4_f32(
            false, a, false, b0, (short)0, acc0, false, false);
        v2f b1 = *(const v2f*)(Bb + 16 * APAD);
        acc1 = __builtin_amdgcn_wmma_f32_16x16x4_f32(
            false, a, false, b1, (short)0, acc1, false, false);
        v2f b2 = *(const v2f*)(Bb + 32 * APAD);
        acc2 = __builtin_amdgcn_wmma_f32_16x16x4_f32(
            false, a, false, b2, (short)0, acc2, false, false);
    }

    // epilogue: restage through LDS so global stores are contiguous along C
    __syncthreads();
    float* so = sB;                           // reuse B staging (48 x 130 fits)
    #pragma unroll
    for (int v = 0; v < 8; ++v) {
        int row = wave * 16 + v + 8 * hi;     // c_local of D element
        so[(0 * 16 + l15) * OPAD + row] = acc0[v];
        so[(1 * 16 + l15) * OPAD + row] = acc1[v];
        so[(2 * 16 + l15) * OPAD + row] = acc2[v];
    }
    __syncthreads();
    #pragma unroll
    for (int i = 0; i < 24; ++i) {
        int idx = t + i * 256;                // 0..6143
        int p = idx >> 7, cc = idx & 127;
        out[((size_t)(b * PRED + p0 + p)) * NC + c0 + cc] = so[p * OPAD + cc];
    }
}

// ---------------------------------------------------------------------------
extern "C" void kernel_launch(void* const* d_in, const int* in_sizes, int n_in,
                              void* d_out, int out_size, void* d_ws, size_t ws_size,
                              hipStream_t stream) {
    const float* x   = (const float*)d_in[0];
    const float* WgS = (const float*)d_in[1];
    const float* WeS = (const float*)d_in[2];
    const float* beS = (const float*)d_in[3];
    const float* WgT = (const float*)d_in[4];
    const float* WeT = (const float*)d_in[5];
    const float* beT = (const float*)d_in[6];
    float* out = (float*)d_out;
    float* ws  = (float*)d_ws;

    float* sea   = ws + WS_SEA;
    float* trend = ws + WS_TREND;
    float* wfS   = ws + WS_WFS;
    float* wfT   = ws + WS_WFT;
    float* gate  = ws + WS_GATE;

    k_decompose<<<dim3(2, NB), 256, 0, stream>>>(x, sea, trend);
    k_gate     <<<dim3(2, NB), 256, 0, stream>>>(sea, trend, WgS, WgT, wfS, wfT, gate);
    k_gatemean <<<16, 256, 0, stream>>>(gate, out + (size_t)NB * PRED * NC);
    k_moe      <<<dim3(PRED / 48, NR / 128), 256, 0, stream>>>(
        sea, trend, WeS, WeT, beS, beT, wfS, wfT, out);
}